// Feature_Select_Layer_23733989277985
// MI455X (gfx1250) — compile-verified
//
#include <hip/hip_runtime.h>
#include <stdint.h>

typedef __attribute__((ext_vector_type(4))) float v4f;
typedef __attribute__((ext_vector_type(4))) int   v4i;

#ifndef __has_builtin
#define __has_builtin(x) 0
#endif

#if __has_builtin(__builtin_amdgcn_global_load_async_to_lds_b128) && \
    __has_builtin(__builtin_amdgcn_s_wait_asynccnt)
#define FSL_ASYNC_LDS 1
#else
#define FSL_ASYNC_LDS 0
#endif

// Address-space-qualified pointer types for the async-LDS builtin
// (clang prints AS1 as __device__, AS3 as __shared__ in HIP mode).
typedef __attribute__((address_space(1))) v4i* gv4i_p;
typedef __attribute__((address_space(3))) v4i* lv4i_p;

// ---------------------------------------------------------------------------
// Async (or fallback) 16-byte global -> LDS copy
// ---------------------------------------------------------------------------
__device__ __forceinline__ void fsl_copy_b128_to_lds(const float* g, float* l) {
#if FSL_ASYNC_LDS
    __builtin_amdgcn_global_load_async_to_lds_b128(
        (gv4i_p)g, (lv4i_p)l, /*offset=*/0, /*cpol=*/0);
#else
    *(v4f*)l = *(const v4f*)g;
#endif
}

__device__ __forceinline__ void fsl_wait_async() {
#if FSL_ASYNC_LDS
    __builtin_amdgcn_s_wait_asynccnt(0);
#endif
}

// ---------------------------------------------------------------------------
// Kernel 1: counting selection. Grid = (D/128)*8 blocks, 128 threads.
//   blockIdx = eg*8 + sg : element-group eg (128 elems), slice-group sg (D/8).
//   Each block stages its 512-float slice of `kernel` into LDS (async B128),
//   each thread counts how many slice values are strictly greater than its
//   element, and atomically accumulates into counts[i].
// ---------------------------------------------------------------------------
#define FSL_SLICES   8
#define FSL_CTHREADS 128

__global__ __launch_bounds__(FSL_CTHREADS) void fsl_count_kernel(
    const float* __restrict__ kern, unsigned* __restrict__ counts, int D) {
    const int slice = D / FSL_SLICES;             // 512 for D=4096
    __shared__ float s_slice[512];
    const int sg  = blockIdx.x & (FSL_SLICES - 1);
    const int eg  = blockIdx.x >> 3;
    const int tid = threadIdx.x;

    // Stage slice: 128 threads * 16B = 2048B = 512 floats.
    if (tid * 4 + 3 < slice) {
        fsl_copy_b128_to_lds(kern + sg * slice + tid * 4, &s_slice[tid * 4]);
    }
    fsl_wait_async();
    __syncthreads();

    const int i = eg * FSL_CTHREADS + tid;
    const float v = kern[i];

    int cnt = 0;
    const v4f* sp = (const v4f*)s_slice;
    const int n4 = slice >> 2;                    // 128
#pragma unroll 4
    for (int j = 0; j < n4; ++j) {
        v4f s = sp[j];                            // LDS broadcast, no conflicts
        cnt += (s.x > v) + (s.y > v) + (s.z > v) + (s.w > v);
    }
    atomicAdd(&counts[i], (unsigned)cnt);
}

// ---------------------------------------------------------------------------
// Kernel 2: build masked kernel. keep element iff (#strictly greater) < k.
// ---------------------------------------------------------------------------
__global__ __launch_bounds__(256) void fsl_finalize_kernel(
    const float* __restrict__ kern, const unsigned* __restrict__ counts,
    const int* __restrict__ kptr, float* __restrict__ ksel, int D) {
    const int i = blockIdx.x * 256 + threadIdx.x;
    if (i < D) {
        const unsigned k = (unsigned)*kptr;
        ksel[i] = (counts[i] < k) ? kern[i] : 0.0f;
    }
}

// ---------------------------------------------------------------------------
// Kernel 3: streaming multiply. Each thread owns one float4-column and 8
// consecutive rows: load ksel once, then 8 NT B128 load/mul/store pairs.
//   c4      = column-block*256 + tid       (float4 column index)
//   rowBase = row-strip * 8
// ---------------------------------------------------------------------------
#define FSL_ROWS_PER_THREAD 8

__global__ __launch_bounds__(256) void fsl_mul_kernel(
    const float* __restrict__ x, const float* __restrict__ ksel,
    float* __restrict__ out, int D4, int colBlocks) {
    const int cb      = blockIdx.x % colBlocks;
    const int rs      = blockIdx.x / colBlocks;
    const int c4      = cb * 256 + threadIdx.x;
    const int rowBase = rs * FSL_ROWS_PER_THREAD;

    const v4f kv = ((const v4f*)ksel)[c4];        // cached (L0/L2), loaded once

    const v4f* xp = (const v4f*)x  + (size_t)rowBase * D4 + c4;
    v4f*       op = (v4f*)out      + (size_t)rowBase * D4 + c4;

#pragma unroll
    for (int r = 0; r < FSL_ROWS_PER_THREAD; ++r) {
        v4f xv = __builtin_nontemporal_load(xp + (size_t)r * D4);
        __builtin_nontemporal_store(xv * kv, op + (size_t)r * D4);
    }
}

// ---------------------------------------------------------------------------
// Launch
// ---------------------------------------------------------------------------
extern "C" void kernel_launch(void* const* d_in, const int* in_sizes, int n_in,
                              void* d_out, int out_size, void* d_ws, size_t ws_size,
                              hipStream_t stream) {
    const float* x    = (const float*)d_in[0];
    const float* kern = (const float*)d_in[1];
    const int*   kptr = (const int*)d_in[2];
    float*       out  = (float*)d_out;

    const int D = in_sizes[1];          // 4096 (kernel length)
    const int B = in_sizes[0] / D;      // 16384 rows

    // Workspace layout: [0, D) uint counts | [D, 2D) float masked kernel
    unsigned* counts = (unsigned*)d_ws;
    float*    ksel   = (float*)((char*)d_ws + (size_t)D * sizeof(unsigned));

    (void)hipMemsetAsync(counts, 0, (size_t)D * sizeof(unsigned), stream);

    // 1) counting selection: (D/128) element-groups x 8 slice-groups
    fsl_count_kernel<<<dim3((D / FSL_CTHREADS) * FSL_SLICES),
                       dim3(FSL_CTHREADS), 0, stream>>>(kern, counts, D);

    // 2) masked kernel vector
    fsl_finalize_kernel<<<dim3((D + 255) / 256), dim3(256), 0, stream>>>(
        kern, counts, kptr, ksel, D);

    // 3) bandwidth-bound broadcast multiply
    const int D4        = D / 4;                       // 1024 float4 columns
    const int colBlocks = D4 / 256;                    // 4
    const int rowStrips = B / FSL_ROWS_PER_THREAD;     // 2048
    fsl_mul_kernel<<<dim3(colBlocks * rowStrips), dim3(256), 0, stream>>>(
        x, ksel, out, D4, colBlocks);
}